// M4_86749749444857
// MI455X (gfx1250) — compile-verified
//
#include <hip/hip_runtime.h>
#include <stdint.h>

// Problem constants (match reference setup_inputs()).
#define N_USERS 100000
#define N_ITEMS 50000
#define N_NODES (N_USERS + N_ITEMS)     // 150000
#define DIM 64
#define NNZ 4800000
#define ND (N_NODES * DIM)              // 9,600,000 floats = 38.4 MB

// SpMM tiling: one tile = 2 edges; 16 lanes per edge, 16B (float4) per lane.
#define NTILES (NNZ / 2)                // 2,400,000
#define DEPTH 4                         // async pipeline depth (tiles in flight)
#define WAVES_PER_BLOCK 8
#define SPMM_BLOCK 256
#define SPMM_GRID 2048

// ---------------------------------------------------------------------------
// Zero three ND-float buffers (hop accumulators) with float4 stores.
// ---------------------------------------------------------------------------
__global__ __launch_bounds__(256) void zero3_kernel(float* __restrict__ a,
                                                    float* __restrict__ b,
                                                    float* __restrict__ c) {
  const int n4 = ND / 4;
  float4 z; z.x = 0.f; z.y = 0.f; z.z = 0.f; z.w = 0.f;
  int i = blockIdx.x * blockDim.x + threadIdx.x;
  const int stride = gridDim.x * blockDim.x;
  for (int j = i; j < n4; j += stride) {
    reinterpret_cast<float4*>(a)[j] = z;
    reinterpret_cast<float4*>(b)[j] = z;
    reinterpret_cast<float4*>(c)[j] = z;
  }
}

// ---------------------------------------------------------------------------
// COO SpMM:  dst[row[e], :] += vals[e] * X[col[e], :]
// X is addressed as two segments so hop-1 can read user/item tables in place:
//   X[c] = (c < N_USERS) ? srcU + c*DIM : srcI + (c - N_USERS)*DIM
// Gather path: per-lane global_load_async_to_lds_b128 (ASYNCcnt), 4-deep
// software pipeline per wave. Scatter path: no-return global_atomic_add_f32.
// ---------------------------------------------------------------------------
__global__ __launch_bounds__(SPMM_BLOCK) void spmm_async_kernel(
    const float* __restrict__ srcU, const float* __restrict__ srcI,
    const int* __restrict__ row, const int* __restrict__ col,
    const float* __restrict__ vals, float* __restrict__ dst) {
  // 8 waves * 4 slots * 2 edges * 64 floats * 4B = 16 KB LDS
  __shared__ __align__(16) float sh[WAVES_PER_BLOCK][DEPTH][2][DIM];

  const int lane = threadIdx.x & 31;
  const int wave = threadIdx.x >> 5;
  const int half = lane >> 4;   // which of the 2 edges in this tile
  const int sub  = lane & 15;   // which 16B chunk of the 64-float row
  const int wstride = gridDim.x * WAVES_PER_BLOCK;          // tile stride
  const int tile0 = blockIdx.x * WAVES_PER_BLOCK + wave;

  int   rreg[DEPTH];
  float vreg[DEPTH];
  uint32_t lds_addr[DEPTH];
#pragma unroll
  for (int s = 0; s < DEPTH; ++s)
    lds_addr[s] = (uint32_t)(uintptr_t)(&sh[wave][s][half][sub * 4]);

  // Issue the gather for `tile` into pipeline slot s. `tile` is wave-uniform,
  // so ASYNCcnt accounting is uniform. Tiles past the end issue a clamped
  // dummy load (never consumed) to keep the in-flight count invariant.
  auto issue = [&](int tile, int s) {
    const int t = (tile < NTILES) ? tile : (NTILES - 1);
    const int e = 2 * t + half;
    const int c = col[e];
    rreg[s] = row[e];
    vreg[s] = vals[e];
    const float* p = (c < N_USERS) ? (srcU + (size_t)c * DIM)
                                   : (srcI + (size_t)(c - N_USERS) * DIM);
    p += sub * 4;
    asm volatile("global_load_async_to_lds_b128 %0, %1, off"
                 :
                 : "v"(lds_addr[s]), "v"((uint64_t)(uintptr_t)p)
                 : "memory");
  };

  // Prologue: fill the pipeline (ASYNCcnt = 4).
#pragma unroll
  for (int s = 0; s < DEPTH; ++s) issue(tile0 + s * wstride, s);

  for (int T = tile0; T < NTILES; T += DEPTH * wstride) {
#pragma unroll
    for (int s = 0; s < DEPTH; ++s) {
      const int myT = T + s * wstride;
      if (myT >= NTILES) break;  // uniform tail exit
      // Oldest outstanding async load (this slot) is complete when <= 3 remain
      // (async loads complete in order).
      asm volatile("s_wait_asynccnt 0x3" ::: "memory");
      const float4 x =
          *reinterpret_cast<const float4*>(&sh[wave][s][half][sub * 4]);
      const float v = vreg[s];
      float* dp = dst + (size_t)rreg[s] * DIM + sub * 4;
      const uint64_t da = (uint64_t)(uintptr_t)dp;
      asm volatile("global_atomic_add_f32 %0, %1, off"
                   :: "v"(da), "v"(x.x * v) : "memory");
      asm volatile("global_atomic_add_f32 %0, %1, off offset:4"
                   :: "v"(da), "v"(x.y * v) : "memory");
      asm volatile("global_atomic_add_f32 %0, %1, off offset:8"
                   :: "v"(da), "v"(x.z * v) : "memory");
      asm volatile("global_atomic_add_f32 %0, %1, off offset:12"
                   :: "v"(da), "v"(x.w * v) : "memory");
      // Refill this slot (keeps 4 in flight; dummies in the tail are fine,
      // s_endpgm's implicit wait-idle drains them).
      issue(myT + DEPTH * wstride, s);
    }
  }
}

// ---------------------------------------------------------------------------
// all_embeddings = (Y1 + Y2 + Y3) / 3  (Y1 already lives in d_out[ND:2ND]).
// ---------------------------------------------------------------------------
__global__ __launch_bounds__(256) void finalize_kernel(
    const float* __restrict__ L, const float* __restrict__ B,
    const float* __restrict__ C, float* __restrict__ out) {
  const int n4 = ND / 4;
  const float k = 1.0f / 3.0f;
  int i = blockIdx.x * blockDim.x + threadIdx.x;
  const int stride = gridDim.x * blockDim.x;
  for (int j = i; j < n4; j += stride) {
    float4 a = reinterpret_cast<const float4*>(L)[j];
    float4 b = reinterpret_cast<const float4*>(B)[j];
    float4 c = reinterpret_cast<const float4*>(C)[j];
    float4 o;
    o.x = (a.x + b.x + c.x) * k;
    o.y = (a.y + b.y + c.y) * k;
    o.z = (a.z + b.z + c.z) * k;
    o.w = (a.w + b.w + c.w) * k;
    reinterpret_cast<float4*>(out)[j] = o;
  }
}

// ---------------------------------------------------------------------------
extern "C" void kernel_launch(void* const* d_in, const int* in_sizes, int n_in,
                              void* d_out, int out_size, void* d_ws,
                              size_t ws_size, hipStream_t stream) {
  (void)in_sizes; (void)n_in; (void)out_size; (void)ws_size;
  const float* user_embed = (const float*)d_in[0];
  const float* item_embed = (const float*)d_in[1];
  const int*   row        = (const int*)d_in[2];
  const int*   col        = (const int*)d_in[3];
  const float* vals       = (const float*)d_in[4];

  float* out   = (float*)d_out;
  float* layer = out + ND;            // hop-1 result == layer_embeddings output
  float* Y2    = (float*)d_ws;        // hop-2 result (38.4 MB)
  float* Y3    = Y2 + ND;             // hop-3 result (38.4 MB)

  zero3_kernel<<<2048, 256, 0, stream>>>(layer, Y2, Y3);

  // hop 1: X = [user_embed ; item_embed]
  spmm_async_kernel<<<SPMM_GRID, SPMM_BLOCK, 0, stream>>>(
      user_embed, item_embed, row, col, vals, layer);
  // hop 2: X = layer (contiguous), split addressing degenerates to one buffer
  spmm_async_kernel<<<SPMM_GRID, SPMM_BLOCK, 0, stream>>>(
      layer, layer + (size_t)N_USERS * DIM, row, col, vals, Y2);
  // hop 3
  spmm_async_kernel<<<SPMM_GRID, SPMM_BLOCK, 0, stream>>>(
      Y2, Y2 + (size_t)N_USERS * DIM, row, col, vals, Y3);

  finalize_kernel<<<2048, 256, 0, stream>>>(layer, Y2, Y3, out);
}